// BoundaryLoss_51256139710654
// MI455X (gfx1250) — compile-verified
//
#include <hip/hip_runtime.h>
#include <hip/hip_bf16.h>

// BoundaryLoss on MI455X (gfx1250, wave32).
// Pipeline (all intermediates in d_ws, everything f32):
//   1. init: zero the two max accumulators.
//   2. row_pass: per-row exact nearest-background distance (two chunked
//      cummax scans in LDS), writes g^2 for both images.
//   3. col_pass x2: brute-force min-plus D2[i,j] = min_k (i-k)^2 + g2[k,j].
//      g2 tiles (64x32 f32) are streamed into LDS by the Tensor Data Mover
//      (tensor_load_to_lds, double-buffered, s_wait_tensorcnt) while all 8
//      waves run the fmin/FMA inner loop.
//   4. max_reduce x2: max(D2) via shuffle + atomicMax on IEEE bits (exact,
//      order-independent -> deterministic).
//   5. final_pass: normalized dists, boundary mask, masked |diff| and count;
//      wave-wide sums via V_WMMA_F32_16X16X4_F32 (B = ones => row sums),
//      deterministic fixed-slot block partials.
//   6. finalize: fixed-order reduction of 1024 partials (WMMA wave sums),
//      out = sum_diff / max(count, 1).

#define HH 1024
#define WW 1024
#define NN (HH * WW)
#define BIGD 1.0e4f
#define SENT 1.0e9f
#define NEGINF (-3.0e38f)

typedef __attribute__((ext_vector_type(2))) float v2f;
typedef __attribute__((ext_vector_type(8))) float v8f;
typedef __attribute__((ext_vector_type(4))) unsigned int u32x4;
typedef __attribute__((ext_vector_type(8))) int i32x8;
typedef __attribute__((ext_vector_type(4))) int i32x4;

// ---------------------------------------------------------------- WMMA wave sum
// A(16x4) layout: lanes 0-15 give A[lane, 0..1] in a.x/a.y, lanes 16-31 give
// A[lane-16, 2..3]. With a = {v, 0} and B = ones(4x16):
//   D[m, n] = v(lane m) + v(lane m + 16)  (replicated across n)
// C/D layout: lane L<16 holds rows 0..7 in c[0..7]; lane L>=16 holds rows 8..15.
// => per-lane s = sum(c[0..7]); wave total = s@lane0 + s@lane16.
__device__ __forceinline__ float wave_sum_wmma(float v) {
    v2f a; a.x = v;    a.y = 0.0f;
    v2f b; b.x = 1.0f; b.y = 1.0f;
    v8f c = {};
    c = __builtin_amdgcn_wmma_f32_16x16x4_f32(false, a, false, b,
                                              (short)0, c, false, false);
    float s = ((c[0] + c[1]) + (c[2] + c[3])) + ((c[4] + c[5]) + (c[6] + c[7]));
    return __shfl(s, 0, 32) + __shfl(s, 16, 32);
}

// ---------------------------------------------------------------- TDM tile load
// 2-D D# per CDNA5 ISA: load a TKxTJ f32 tile whose
// first element is at `gsrc` from a row-major tensor with row stride WW
// into LDS at byte offset `lds_byte`. Issued per-wave; tracked by TENSORcnt.
// This toolchain exposes the 6-arg builtin:
//   (uint32x4 g0, int32x8 g1, int32x4 g2, int32x4 g3, int32x8 g?, i32 cpol)
#define TI 32
#define TJ 32
#define TK 64
__device__ __forceinline__ void tdm_load_tile(const float* gsrc, unsigned lds_byte) {
    unsigned long long ga = (unsigned long long)gsrc;
    u32x4 g0;
    g0[0] = 1u;                                   // count=1, user D#, no gather
    g0[1] = lds_byte;                             // lds_addr [63:32]
    g0[2] = (unsigned)(ga & 0xffffffffu);         // global_addr [95:64]
    g0[3] = (unsigned)((ga >> 32) & 0x01ffffffu)  // global_addr [120:96]
            | (2u << 30);                         // type=2 ("image")
    i32x8 g1;
    g1[0] = (int)(2u << 16);                      // data_size=4B, wg_mask=0
    g1[1] = (int)((unsigned)WW << 16);            // tensor_dim0 = 1024
    g1[2] = (int)((unsigned)HH << 16);            // tensor_dim1 = 1024
    g1[3] = (int)((unsigned)TJ << 16);            // tile_dim0 = 32 (x)
    g1[4] = (int)TK;                              // tile_dim1 = 64 (y), tile_dim2=0
    g1[5] = (int)WW;                              // tensor_dim0_stride = 1024
    g1[6] = 0;                                    // stride0 hi / stride1 lo
    g1[7] = 0;                                    // stride1 hi
    i32x4 z4 = {0, 0, 0, 0};                      // groups 2/3 unused (2-D tile)
    i32x8 z8 = {0, 0, 0, 0, 0, 0, 0, 0};
    __builtin_amdgcn_tensor_load_to_lds(g0, g1, z4, z4, z8, 0);
}

// ---------------------------------------------------------------- init
__global__ void bl_init(unsigned* scal) {
    if (threadIdx.x < 2) scal[threadIdx.x] = 0u;   // m2_gt, m2_pred bits
}

// ---------------------------------------------------------------- row pass
// Inclusive block-wide cummax over 1024 LDS floats, 256 threads, 4 elems each.
__device__ void block_cummax(float* s, float* agg) {
    const int t = threadIdx.x;
    float a0 = s[4 * t + 0];
    float a1 = fmaxf(a0, s[4 * t + 1]);
    float a2 = fmaxf(a1, s[4 * t + 2]);
    float a3 = fmaxf(a2, s[4 * t + 3]);
    agg[t] = a3;
    __syncthreads();
#pragma unroll
    for (int off = 1; off < 256; off <<= 1) {
        float v = agg[t];
        float u = (t >= off) ? agg[t - off] : NEGINF;
        __syncthreads();
        agg[t] = fmaxf(v, u);
        __syncthreads();
    }
    float pre = (t > 0) ? agg[t - 1] : NEGINF;
    s[4 * t + 0] = fmaxf(a0, pre);
    s[4 * t + 1] = fmaxf(a1, pre);
    s[4 * t + 2] = fmaxf(a2, pre);
    s[4 * t + 3] = fmaxf(a3, pre);
    __syncthreads();
}

__global__ void row_pass(const float* __restrict__ gt,
                         const float* __restrict__ pred,
                         float* __restrict__ g2_gt,
                         float* __restrict__ g2_pred) {
    const int row = blockIdx.x;
    const int t   = threadIdx.x;          // 256 threads
    __shared__ float sv[WW];
    __shared__ float agg[256];

    for (int img = 0; img < 2; ++img) {
        const float* src = (img == 0) ? gt : pred;
        float* dst       = (img == 0) ? g2_gt : g2_pred;
        bool z[4];
#pragma unroll
        for (int q = 0; q < 4; ++q) {
            int j = 4 * t + q;
            float x = src[row * WW + j];
            // background: gt == 0 ; sigmoid(pred) <= 0.5  <=>  pred <= 0
            z[q] = (img == 0) ? (x == 0.0f) : (x <= 0.0f);
            sv[j] = z[q] ? (float)j : -SENT;
        }
        __syncthreads();
        block_cummax(sv, agg);            // nearest background to the left
        float leftv[4];
#pragma unroll
        for (int q = 0; q < 4; ++q) leftv[q] = sv[4 * t + q];
        __syncthreads();
#pragma unroll
        for (int q = 0; q < 4; ++q) {     // reversed scan for the right side
            int j = 4 * t + q;
            int r = WW - 1 - j;
            sv[r] = z[q] ? (float)r : -SENT;
        }
        __syncthreads();
        block_cummax(sv, agg);
        float g2v[4];
#pragma unroll
        for (int q = 0; q < 4; ++q) {
            int j = 4 * t + q;
            float rightv = (float)(WW - 1) - sv[WW - 1 - j];
            float g = fminf(fminf((float)j - leftv[q], rightv - (float)j), BIGD);
            g2v[q] = g * g;
        }
        *(float4*)&dst[row * WW + 4 * t] =
            make_float4(g2v[0], g2v[1], g2v[2], g2v[3]);
        __syncthreads();
    }
}

// ---------------------------------------------------------------- column min-plus
// Double-buffered TDM staging: wave 0 waits TENSORcnt, the workgroup barrier
// publishes LDS, wave 0 issues the next chunk, all waves compute on `cur`.
__global__ void col_pass(const float* __restrict__ g2, float* __restrict__ D2) {
    const int jt  = blockIdx.x * TJ;
    const int it  = blockIdx.y * TI;
    const int t   = threadIdx.x;          // 256
    const int lj  = t & (TJ - 1);
    const int li0 = t >> 5;               // 0..7
    __shared__ float tile[2][TK][TJ];     // 16 KB double buffer

    if (t < 32)                           // wave 0 launches chunk 0
        tdm_load_tile(&g2[0 * WW + jt],
                      (unsigned)(unsigned long long)&tile[0][0][0]);

    float best[4] = {3.0e38f, 3.0e38f, 3.0e38f, 3.0e38f};
    int cur = 0;
    for (int kb = 0; kb < HH; kb += TK) {
        if (t < 32) __builtin_amdgcn_s_wait_tensorcnt(0);
        __syncthreads();                  // publish chunk `cur` to all waves
        if (t < 32 && (kb + TK) < HH)     // prefetch chunk kb+TK into buf^1
            tdm_load_tile(&g2[(kb + TK) * WW + jt],
                          (unsigned)(unsigned long long)&tile[cur ^ 1][0][0]);
#pragma unroll 8
        for (int k = 0; k < TK; ++k) {
            float gv = tile[cur][k][lj];
            float kk = (float)(kb + k);
#pragma unroll
            for (int q = 0; q < 4; ++q) {
                float di = (float)(it + li0 + 8 * q) - kk;
                best[q] = fminf(best[q], __builtin_fmaf(di, di, gv));
            }
        }
        __syncthreads();                  // buf^1 read-done before its overwrite
        cur ^= 1;
    }
#pragma unroll
    for (int q = 0; q < 4; ++q)
        D2[(it + li0 + 8 * q) * WW + jt + lj] = best[q];
}

// ---------------------------------------------------------------- max reduce
__global__ void max_reduce(const float* __restrict__ D2, unsigned* out_bits) {
    const int tid = blockIdx.x * blockDim.x + threadIdx.x;    // 262144 threads
    float4 v = *(const float4*)&D2[tid * 4];
    float m = fmaxf(fmaxf(v.x, v.y), fmaxf(v.z, v.w));        // D2 >= 0
#pragma unroll
    for (int off = 16; off > 0; off >>= 1)
        m = fmaxf(m, __shfl_xor(m, off, 32));
    if ((threadIdx.x & 31) == 0)
        atomicMax(out_bits, __float_as_uint(m));              // exact for f32>=0
}

// ---------------------------------------------------------------- masked sums
__global__ void final_pass(const float* __restrict__ D2g,
                           const float* __restrict__ D2p,
                           const unsigned* __restrict__ scal,
                           float* __restrict__ psd, float* __restrict__ psc) {
    const int t   = threadIdx.x;          // 256
    const int tid = blockIdx.x * 256 + t; // 1024 blocks
    const float invMg = 1.0f / (sqrtf(__uint_as_float(scal[0])) + 1e-6f);
    const float invMp = 1.0f / (sqrtf(__uint_as_float(scal[1])) + 1e-6f);

    float4 a = *(const float4*)&D2g[tid * 4];
    float4 b = *(const float4*)&D2p[tid * 4];
    float sd = 0.0f, sc = 0.0f;
    float ga[4] = {a.x, a.y, a.z, a.w};
    float pa[4] = {b.x, b.y, b.z, b.w};
#pragma unroll
    for (int q = 0; q < 4; ++q) {
        float gd = sqrtf(ga[q]) * invMg;
        float pd = sqrtf(pa[q]) * invMp;
        bool msk = (gd < 0.1f) || (pd < 0.1f);
        sd += msk ? fabsf(gd - pd) : 0.0f;
        sc += msk ? 1.0f : 0.0f;
    }
    // EXEC is all ones here (no divergence): WMMA wave reduction.
    float wsd = wave_sum_wmma(sd);
    float wsc = wave_sum_wmma(sc);
    __shared__ float lsd[8], lsc[8];
    if ((t & 31) == 0) { lsd[t >> 5] = wsd; lsc[t >> 5] = wsc; }
    __syncthreads();
    if (t == 0) {
        float A = 0.0f, B = 0.0f;
#pragma unroll
        for (int w = 0; w < 8; ++w) { A += lsd[w]; B += lsc[w]; }
        psd[blockIdx.x] = A;              // fixed slot -> deterministic
        psc[blockIdx.x] = B;
    }
}

// ---------------------------------------------------------------- finalize
__global__ void finalize(const float* __restrict__ psd,
                         const float* __restrict__ psc,
                         float* __restrict__ out) {
    const int t = threadIdx.x;            // 256 threads, 1 block
    float sd = ((psd[t] + psd[t + 256]) + psd[t + 512]) + psd[t + 768];
    float sc = ((psc[t] + psc[t + 256]) + psc[t + 512]) + psc[t + 768];
    float wsd = wave_sum_wmma(sd);
    float wsc = wave_sum_wmma(sc);
    __shared__ float lsd[8], lsc[8];
    if ((t & 31) == 0) { lsd[t >> 5] = wsd; lsc[t >> 5] = wsc; }
    __syncthreads();
    if (t == 0) {
        float A = 0.0f, B = 0.0f;
#pragma unroll
        for (int w = 0; w < 8; ++w) { A += lsd[w]; B += lsc[w]; }
        out[0] = A / fmaxf(B, 1.0f);
    }
}

// ---------------------------------------------------------------- launch
extern "C" void kernel_launch(void* const* d_in, const int* in_sizes, int n_in,
                              void* d_out, int out_size, void* d_ws, size_t ws_size,
                              hipStream_t stream) {
    const float* pred = (const float*)d_in[0];
    const float* gt   = (const float*)d_in[1];
    float* out        = (float*)d_out;

    float* ws = (float*)d_ws;             // needs ~16.05 MB of scratch
    float* g2_gt   = ws;                  // 1M f32
    float* g2_pred = ws + (1 << 20);
    float* D2_gt   = ws + (2 << 20);
    float* D2_pred = ws + (3 << 20);
    unsigned* scal = (unsigned*)(ws + (4 << 20));          // [0]=m2_gt [1]=m2_pred
    float* psd     = ws + (4 << 20) + 64;                  // 1024 block partials
    float* psc     = ws + (4 << 20) + 64 + 1024;

    bl_init<<<1, 32, 0, stream>>>(scal);
    row_pass<<<HH, 256, 0, stream>>>(gt, pred, g2_gt, g2_pred);
    col_pass<<<dim3(WW / TJ, HH / TI), 256, 0, stream>>>(g2_gt, D2_gt);
    col_pass<<<dim3(WW / TJ, HH / TI), 256, 0, stream>>>(g2_pred, D2_pred);
    max_reduce<<<NN / (256 * 4), 256, 0, stream>>>(D2_gt, &scal[0]);
    max_reduce<<<NN / (256 * 4), 256, 0, stream>>>(D2_pred, &scal[1]);
    final_pass<<<NN / (256 * 4), 256, 0, stream>>>(D2_gt, D2_pred, scal, psd, psc);
    finalize<<<1, 256, 0, stream>>>(psd, psc, out);
}